// Covid19InfectionsPredictModel_15126874816739
// MI455X (gfx1250) — compile-verified
//
#include <hip/hip_runtime.h>
#include <math.h>

// ---------------------------------------------------------------------------
// Covid19 infections model, CDNA5 (gfx1250, wave32).
//
//   rate = relu(inputs @ a)                      -> WMMA f32 16x16x4 GEMV
//   x_t  = rate_t * dot(window_t, rev_head)      -> linear time-varying scan,
//   out_t = dot(window_t, diff)                     parallelized by 20x20
//                                                   block-transition matrices
// ---------------------------------------------------------------------------

typedef __attribute__((ext_vector_type(2))) float v2f;
typedef __attribute__((ext_vector_type(8))) float v8f;

#define T_TOTAL   65536
#define NW        64
#define LM        20      // window length (MAX_LATENCY - 1)
#define TBLK      256     // timesteps per scan block
#define NBLK      (T_TOTAL / TBLK)

// --- shared coefficient helper (rev_head & diff from h) ---------------------
// s[t] = (exp(-t/h) - exp(-20/h)) / (1 - exp(-20/h));  s[20] == 0 exactly.
// rev_head[i] = s[20-i]   (i = 0..19)
// diff[i]     = s[19-i] - s[20-i]
__device__ __forceinline__ void compute_coeffs(float h, float* rh, float* df) {
    float e20   = expf(-20.0f / h);
    float denom = 1.0f - e20;
    #pragma unroll
    for (int i = 0; i < LM; ++i) {
        float s_hi = (i == 0) ? 0.0f
                              : (expf(-(float)(20 - i) / h) - e20) / denom; // s[20-i]
        float s_lo = (expf(-(float)(19 - i) / h) - e20) / denom;            // s[19-i]
        rh[i] = s_hi;
        df[i] = s_lo - s_hi;
    }
}

// ---------------------------------------------------------------------------
// Kernel 1: rate = relu(inputs @ a) via V_WMMA_F32_16X16X4_F32.
// One wave handles 16 rows. B fragment replicates `a` across all 16 columns,
// so every column of D equals the row dot-product.
// A 16x4 f32 layout: lanes 0-15 hold M=lane, VGPR0/1 = K0/K1; lanes 16-31
// hold K2/K3 (ISA 7.12.2) -> one aligned float2 load per lane per K-chunk.
// ---------------------------------------------------------------------------
__global__ void rate_wmma_kernel(const float* __restrict__ in,
                                 const float* __restrict__ a,
                                 float* __restrict__ rate) {
    const int lane = threadIdx.x & 31;
    const int wave = blockIdx.x * (blockDim.x >> 5) + (threadIdx.x >> 5);
    const int row0 = wave * 16;
    const int m    = lane & 15;     // row within tile
    const int hi   = lane >> 4;     // 0: K{0,1},  1: K{2,3}

    const float* rowp = in + (size_t)(row0 + m) * NW;

    v8f c = {};
    #pragma unroll
    for (int k = 0; k < NW; k += 4) {
        const int koff = k + 2 * hi;
        v2f av = *(const v2f*)(rowp + koff);   // A[m][koff], A[m][koff+1]
        v2f bv = *(const v2f*)(a    + koff);   // B[koff][n] = a[koff]  (all n)
        c = __builtin_amdgcn_wmma_f32_16x16x4_f32(
                /*neg_a=*/false, av, /*neg_b=*/false, bv,
                /*c_mod=*/(short)0, c, /*reuse_a=*/false, /*reuse_b=*/false);
    }

    // D layout: lanes 0-15 -> M = vgpr(0..7); lanes 16-31 -> M = vgpr+8.
    // Column N is arbitrary (all identical); use N==0 lanes (0 and 16).
    if (m == 0) {
        #pragma unroll
        for (int i = 0; i < 8; ++i) {
            float v = c[i];
            rate[row0 + 8 * hi + i] = v > 0.0f ? v : 0.0f;
        }
    }
}

// ---------------------------------------------------------------------------
// Kernel 2: per-block 20x20 transition matrix P_b = A_{last} ... A_{first}.
// Lane j carries column j of P in registers. Step: shift up, last element =
// r_t * dot(rev_head, column).   P stored row-major: P[b][i][j].
// ---------------------------------------------------------------------------
__global__ void transition_kernel(const float* __restrict__ rate,
                                  const float* __restrict__ hptr,
                                  float* __restrict__ P) {
    __shared__ float rlds[TBLK];
    const int b    = blockIdx.x;
    const int lane = threadIdx.x;   // blockDim.x == 32

    for (int i = lane; i < TBLK; i += 32) rlds[i] = rate[b * TBLK + i];

    float rh[LM], df[LM];
    compute_coeffs(hptr[0], rh, df);
    __syncthreads();

    float p[LM];
    #pragma unroll
    for (int i = 0; i < LM; ++i) p[i] = (i == lane) ? 1.0f : 0.0f;

    for (int t = 0; t < TBLK; ++t) {
        float r   = rlds[t];
        float acc = 0.0f;
        #pragma unroll
        for (int i = 0; i < LM; ++i) acc += rh[i] * p[i];
        float nv = r * acc;
        #pragma unroll
        for (int i = 0; i < LM - 1; ++i) p[i] = p[i + 1];
        p[LM - 1] = nv;
    }

    if (lane < LM) {
        #pragma unroll
        for (int i = 0; i < LM; ++i)
            P[(size_t)b * (LM * LM) + i * LM + lane] = p[i];
    }
}

// ---------------------------------------------------------------------------
// Kernel 3: sequential boundary-window propagation (one wave).
// Wb[b] = window at start of block b.  Wb[b+1] = P_b * Wb[b].
// Lane i owns row i; window broadcast through LDS each iteration.
// ---------------------------------------------------------------------------
__global__ void boundary_kernel(const float* __restrict__ P,
                                float* __restrict__ Wb) {
    __shared__ float wsh[LM];
    const int lane = threadIdx.x;   // 32 threads

    float w = (lane == LM - 1) ? 1.0f : 0.0f;   // W_0 = [0,...,0,1]
    if (lane < LM) Wb[lane] = w;

    for (int b = 0; b < NBLK; ++b) {
        if (lane < LM) wsh[lane] = w;
        __syncthreads();
        float nw = 0.0f;
        if (lane < LM) {
            const float* rowp = P + (size_t)b * (LM * LM) + lane * LM;
            #pragma unroll
            for (int j = 0; j < LM; ++j) nw += rowp[j] * wsh[j];
        }
        __syncthreads();
        w = nw;
        if (lane < LM) Wb[(b + 1) * LM + lane] = w;
    }
}

// ---------------------------------------------------------------------------
// Kernel 4: per-block replay.  Lane 0 runs the 256-step recurrence into LDS
// (critical path ~2 dependent VALU ops per step), then all lanes compute the
// 20-tap FIR outputs  out[t] = dot(window_t, diff)  with coalesced stores.
// ---------------------------------------------------------------------------
__global__ void replay_kernel(const float* __restrict__ rate,
                              const float* __restrict__ hptr,
                              const float* __restrict__ Wb,
                              float* __restrict__ out) {
    __shared__ float x[LM + TBLK];   // x[k] = x_{t0-20+k}
    __shared__ float rlds[TBLK];
    const int b    = blockIdx.x;
    const int lane = threadIdx.x;    // 32 threads

    for (int i = lane; i < TBLK; i += 32) rlds[i] = rate[b * TBLK + i];
    if (lane < LM) x[lane] = Wb[b * LM + lane];

    float rh[LM], df[LM];
    compute_coeffs(hptr[0], rh, df);
    __syncthreads();

    if (lane == 0) {
        float w[LM];
        #pragma unroll
        for (int i = 0; i < LM; ++i) w[i] = x[i];
        for (int t = 0; t < TBLK; ++t) {
            float acc = 0.0f;
            #pragma unroll
            for (int i = 0; i < LM; ++i) acc += rh[i] * w[i];
            float nv = rlds[t] * acc;
            x[LM + t] = nv;
            #pragma unroll
            for (int i = 0; i < LM - 1; ++i) w[i] = w[i + 1];
            w[LM - 1] = nv;
        }
    }
    __syncthreads();

    for (int t = lane; t < TBLK; t += 32) {
        float acc = 0.0f;
        #pragma unroll
        for (int i = 0; i < LM; ++i) acc += x[t + i] * df[i];
        out[b * TBLK + t] = acc;
    }
}

// ---------------------------------------------------------------------------
extern "C" void kernel_launch(void* const* d_in, const int* in_sizes, int n_in,
                              void* d_out, int out_size, void* d_ws, size_t ws_size,
                              hipStream_t stream) {
    const float* in = (const float*)d_in[0];   // [65536, 64]
    const float* a  = (const float*)d_in[1];   // [64]
    const float* h  = (const float*)d_in[2];   // [1]
    float* out = (float*)d_out;                // [65536]

    float* ws   = (float*)d_ws;
    float* rate = ws;                                   // T_TOTAL floats
    float* P    = ws + T_TOTAL;                         // NBLK*400 floats
    float* Wbnd = ws + T_TOTAL + NBLK * (LM * LM);      // (NBLK+1)*20 floats

    // 16 rows per wave, 8 waves per block -> 128 rows per block
    rate_wmma_kernel<<<T_TOTAL / 128, 256, 0, stream>>>(in, a, rate);
    transition_kernel<<<NBLK, 32, 0, stream>>>(rate, h, P);
    boundary_kernel<<<1, 32, 0, stream>>>(P, Wbnd);
    replay_kernel<<<NBLK, 32, 0, stream>>>(rate, h, Wbnd, out);
}